// GCNForMIS_7052336300283
// MI455X (gfx1250) — compile-verified
//
#include <hip/hip_runtime.h>
#include <hip/hip_bf16.h>

typedef __attribute__((ext_vector_type(2))) float v2f;
typedef __attribute__((ext_vector_type(8))) float v8f;

#define TB 256

__device__ __forceinline__ void atomAddF(float* p, float v) {
  // hardware global_atomic_add_f32 (no CAS loop)
  unsafeAtomicAdd(p, v);
}

// ---------- degree / normalization ----------
__global__ void k_init_deg(float* __restrict__ deg, int n) {
  int i = blockIdx.x * blockDim.x + threadIdx.x;
  if (i < n) deg[i] = 1.0f;  // self-loop contribution
}

__global__ void k_count_deg(const long long* __restrict__ col, float* __restrict__ deg, int e) {
  int i = blockIdx.x * blockDim.x + threadIdx.x;
  if (i < e) atomAddF(&deg[col[i]], 1.0f);
}

__global__ void k_dinv(float* __restrict__ deg, int n) {
  int i = blockIdx.x * blockDim.x + threadIdx.x;
  if (i < n) deg[i] = rsqrtf(deg[i]);  // deg >= 1 always
}

// ---------- scalar aggregation (layers 0 and 2) ----------
__global__ void k_init_scalar(const float* __restrict__ x, const float* __restrict__ dinv,
                              float* __restrict__ s, int n) {
  int i = blockIdx.x * blockDim.x + threadIdx.x;
  if (i < n) s[i] = x[i] * dinv[i] * dinv[i];  // self-loop term
}

__global__ void k_init_scalar_bias(const float* __restrict__ x, const float* __restrict__ dinv,
                                   const float* __restrict__ bias, float* __restrict__ s, int n) {
  int i = blockIdx.x * blockDim.x + threadIdx.x;
  if (i < n) s[i] = x[i] * dinv[i] * dinv[i] + bias[0];
}

__global__ void k_edge_scalar(const long long* __restrict__ row, const long long* __restrict__ col,
                              const float* __restrict__ x, const float* __restrict__ dinv,
                              float* __restrict__ s, int e) {
  int i = blockIdx.x * blockDim.x + threadIdx.x;
  if (i >= e) return;
  long long r = row[i], c = col[i];
  atomAddF(&s[c], x[r] * dinv[r] * dinv[c]);
}

// ---------- layer 0: h0 = relu(s0 * W0 + b0), outer product ----------
__global__ void k_h0(const float* __restrict__ s, const float* __restrict__ W0,
                     const float* __restrict__ b0, float* __restrict__ h0, int n) {
  int idx = blockIdx.x * blockDim.x + threadIdx.x;
  int i = idx >> 6, f = idx & 63;
  if (i < n) h0[idx] = fmaxf(s[i] * W0[f] + b0[f], 0.0f);
}

// ---------- layer 1 aggregation: a1 = A_hat_norm @ h0 (64-wide) ----------
__global__ void k_init_vec(const float* __restrict__ h, const float* __restrict__ dinv,
                           float* __restrict__ a, int n) {
  int idx = blockIdx.x * blockDim.x + threadIdx.x;
  int i = idx >> 6;
  if (i < n) a[idx] = h[idx] * dinv[i] * dinv[i];
}

__global__ void k_edge_vec(const long long* __restrict__ row, const long long* __restrict__ col,
                           const float* __restrict__ h, const float* __restrict__ dinv,
                           float* __restrict__ a, int e) {
  int t = blockIdx.x * blockDim.x + threadIdx.x;
  int i = t >> 4;              // 16 lanes per edge, float4 per lane
  if (i >= e) return;
  int f = (t & 15) * 4;
  long long r = row[i], c = col[i];
  float w = dinv[r] * dinv[c];
  const float4 v = *reinterpret_cast<const float4*>(h + r * 64 + f);
  float* dst = a + c * 64 + f;
  atomAddF(dst + 0, v.x * w);
  atomAddF(dst + 1, v.y * w);
  atomAddF(dst + 2, v.z * w);
  atomAddF(dst + 3, v.w * w);
}

// ---------- layer 1 GEMM: h1 = relu(a1 @ W1 + b1), fp32 WMMA ----------
// One wave per 16x16 output tile. V_WMMA_F32_16X16X4_F32, 16 K-steps.
// A frag (16x4): lane L (L<16): M=L, VGPR v holds K=v;     lanes 16-31: K=2+v.
// B frag (4x16): VGPR v, lanes 0-15: K=v, N=lane;          lanes 16-31: K=v+2, N=lane-16.
// D (16x16 f32): VGPR r, lanes 0-15: M=r, N=lane;          lanes 16-31: M=r+8, N=lane-16.
__global__ void k_gemm_wmma(const float* __restrict__ A, const float* __restrict__ W,
                            const float* __restrict__ bias, float* __restrict__ D, int n) {
  int wave = blockIdx.x * (blockDim.x / 32) + (threadIdx.x >> 5);
  int lane = threadIdx.x & 31;
  int rowTile = wave >> 2;        // 64 cols = 4 tiles of 16
  int colTile = wave & 3;
  int rowBase = rowTile * 16;
  if (rowBase >= n) return;       // wave-uniform exit; EXEC stays full for WMMA
  int colBase = colTile * 16;
  int m  = lane & 15;
  int kh = lane >> 4;             // K-half select (0 or 1)

  int arow = rowBase + m;
  if (arow >= n - 1) arow = n - 1;          // clamp loads (stores are guarded)
  const float* ap = A + (long long)arow * 64;

  v8f c = {};
#pragma unroll
  for (int k0 = 0; k0 < 64; k0 += 4) {
    v2f af, bf;
    af.x = ap[k0 + kh * 2 + 0];
    af.y = ap[k0 + kh * 2 + 1];
    bf.x = W[(k0 + kh * 2 + 0) * 64 + colBase + m];
    bf.y = W[(k0 + kh * 2 + 1) * 64 + colBase + m];
    c = __builtin_amdgcn_wmma_f32_16x16x4_f32(false, af, false, bf, (short)0, c,
                                              false, false);
  }

  float bb = bias[colBase + m];
#pragma unroll
  for (int r = 0; r < 8; ++r) {
    int orow = rowBase + r + kh * 8;
    if (orow < n) {
      float v = c[r] + bb;
      D[(long long)orow * 64 + colBase + m] = fmaxf(v, 0.0f);
    }
  }
}

// ---------- layer 2: z = h1 . W_out per node ----------
__global__ void k_dot(const float* __restrict__ h, const float* __restrict__ Wout,
                      float* __restrict__ z, int n) {
  int i = blockIdx.x * blockDim.x + threadIdx.x;
  if (i >= n) return;
  const float4* hp = reinterpret_cast<const float4*>(h + (long long)i * 64);
  const float4* wp = reinterpret_cast<const float4*>(Wout);
  float acc = 0.0f;
#pragma unroll
  for (int q = 0; q < 16; ++q) {
    float4 a = hp[q], b = wp[q];
    acc += a.x * b.x + a.y * b.y + a.z * b.z + a.w * b.w;
  }
  z[i] = acc;
}

__global__ void k_sigmoid(const float* __restrict__ o, float* __restrict__ out, int n) {
  int i = blockIdx.x * blockDim.x + threadIdx.x;
  if (i < n) out[i] = 1.0f / (1.0f + __expf(-o[i]));
}

static inline int gridFor(long long work) { return (int)((work + TB - 1) / TB); }

extern "C" void kernel_launch(void* const* d_in, const int* in_sizes, int n_in,
                              void* d_out, int out_size, void* d_ws, size_t ws_size,
                              hipStream_t stream) {
  const float*     x     = (const float*)d_in[0];
  const long long* ei    = (const long long*)d_in[1];   // int64 per reference
  const float*     W0    = (const float*)d_in[2];
  const float*     b0    = (const float*)d_in[3];
  const float*     W1    = (const float*)d_in[4];
  const float*     b1    = (const float*)d_in[5];
  const float*     W_out = (const float*)d_in[6];
  const float*     b_out = (const float*)d_in[7];
  float*           out   = (float*)d_out;

  const int n = in_sizes[0];          // 100000 nodes
  const int E = in_sizes[1] / 2;      // 3.2M edges
  const long long* row = ei;
  const long long* col = ei + E;

  // workspace layout (floats): dinv[n] | sbuf[n] | obuf[n] | bufA[n*64] | bufB[n*64]
  float* ws   = (float*)d_ws;
  float* dinv = ws;
  float* sbuf = ws + (size_t)n;
  float* obuf = ws + (size_t)2 * n;
  float* bufA = ws + (size_t)3 * n;                    // h0, later h1
  float* bufB = ws + (size_t)3 * n + (size_t)64 * n;   // a1

  // normalization
  k_init_deg  <<<gridFor(n), TB, 0, stream>>>(dinv, n);
  k_count_deg <<<gridFor(E), TB, 0, stream>>>(col, dinv, E);
  k_dinv      <<<gridFor(n), TB, 0, stream>>>(dinv, n);

  // layer 0 (scalar aggregation, then outer product with W0)
  k_init_scalar<<<gridFor(n), TB, 0, stream>>>(x, dinv, sbuf, n);
  k_edge_scalar<<<gridFor(E), TB, 0, stream>>>(row, col, x, dinv, sbuf, E);
  k_h0         <<<gridFor((long long)n * 64), TB, 0, stream>>>(sbuf, W0, b0, bufA, n);

  // layer 1: aggregate h0 (64-wide), then WMMA GEMM with W1 + bias + relu
  k_init_vec <<<gridFor((long long)n * 64), TB, 0, stream>>>(bufA, dinv, bufB, n);
  k_edge_vec <<<gridFor((long long)E * 16), TB, 0, stream>>>(row, col, bufA, dinv, bufB, E);
  {
    int rowTiles = (n + 15) / 16;
    long long waves = (long long)rowTiles * 4;           // 4 col-tiles
    int blocks = (int)((waves + 7) / 8);                 // 8 waves per 256-thread block
    k_gemm_wmma<<<blocks, TB, 0, stream>>>(bufB, W1, b1, bufA, n);
  }

  // layer 2: z = h1 . W_out, then scalar aggregation, then sigmoid
  k_dot            <<<gridFor(n), TB, 0, stream>>>(bufA, W_out, sbuf, n);
  k_init_scalar_bias<<<gridFor(n), TB, 0, stream>>>(sbuf, dinv, b_out, obuf, n);
  k_edge_scalar    <<<gridFor(E), TB, 0, stream>>>(row, col, sbuf, dinv, obuf, E);
  k_sigmoid        <<<gridFor(n), TB, 0, stream>>>(obuf, out, n);
}